// GWeightedSVD_49246095016583
// MI455X (gfx1250) — compile-verified
//
#include <hip/hip_runtime.h>
#include <hip/hip_bf16.h>
#include <math.h>

typedef float v2f __attribute__((ext_vector_type(2)));
typedef float v4f __attribute__((ext_vector_type(4)));
typedef float v8f __attribute__((ext_vector_type(8)));

static constexpr int   kN    = 8192;   // points per batch
static constexpr int   kTPB  = 256;    // 8 waves (wave32)
static constexpr int   kPPT  = 4;      // points per thread per iter (float4)
static constexpr int   kIter = kN / (kTPB * kPPT);  // 8
static constexpr float kEPS  = 0.0001f;

// ---------------------------------------------------------------------------
// Kernel 1: one-pass raw-moment reduction. One block per batch.
// Accumulates 16 scalars: [W, P(3), Q(3), M(9)] where
//   W = sum w, P = sum w*src, Q = sum w*(src+rcs), M_ij = sum w*src_i*cor_j
// Block-level reduction of the 256x16 partials is done with
// V_WMMA_F32_16X16X4_F32 (A = partials 16x4, B = ones) accumulating in f32.
// ---------------------------------------------------------------------------
__global__ __launch_bounds__(kTPB) void moments_kernel(
    const float* __restrict__ src, const float* __restrict__ rcs,
    const float* __restrict__ wts, float* __restrict__ part)
{
    __shared__ float lds[kTPB * 17];   // stride 17 -> conflict-free

    const int b   = blockIdx.x;
    const int tid = threadIdx.x;
    const size_t base3 = (size_t)b * 3u * kN;

    const float* sx = src + base3;
    const float* sy = sx + kN;
    const float* sz = sy + kN;
    const float* rx = rcs + base3;
    const float* ry = rx + kN;
    const float* rz = ry + kN;
    const float* wb = wts + (size_t)b * kN;

    float a[16];
#pragma unroll
    for (int i = 0; i < 16; ++i) a[i] = 0.0f;

    for (int it = 0; it < kIter; ++it) {
        const int idx = it * (kTPB * kPPT) + tid * kPPT;

        // streaming: non-temporal b128 loads (no reuse; dataset > L2)
        v4f vsx = __builtin_nontemporal_load((const v4f*)(sx + idx));
        v4f vsy = __builtin_nontemporal_load((const v4f*)(sy + idx));
        v4f vsz = __builtin_nontemporal_load((const v4f*)(sz + idx));
        v4f vrx = __builtin_nontemporal_load((const v4f*)(rx + idx));
        v4f vry = __builtin_nontemporal_load((const v4f*)(ry + idx));
        v4f vrz = __builtin_nontemporal_load((const v4f*)(rz + idx));
        v4f vw  = __builtin_nontemporal_load((const v4f*)(wb + idx));

        if (it + 1 < kIter) {   // global_prefetch_b8 for the next chunk
            const int nxt = idx + kTPB * kPPT;
            __builtin_prefetch(sx + nxt, 0, 0);
            __builtin_prefetch(wb + nxt, 0, 0);
        }

#pragma unroll
        for (int e = 0; e < 4; ++e) {
            const float w  = vw[e];
            const float x  = vsx[e], y = vsy[e], z = vsz[e];
            const float cx = x + vrx[e];
            const float cy = y + vry[e];
            const float cz = z + vrz[e];
            const float wx = w * x, wy = w * y, wz = w * z;
            a[0]  += w;
            a[1]  += wx;       a[2]  += wy;       a[3]  += wz;
            a[4]  += w * cx;   a[5]  += w * cy;   a[6]  += w * cz;
            a[7]  += wx * cx;  a[8]  += wx * cy;  a[9]  += wx * cz;
            a[10] += wy * cx;  a[11] += wy * cy;  a[12] += wy * cz;
            a[13] += wz * cx;  a[14] += wz * cy;  a[15] += wz * cz;
        }
    }

#pragma unroll
    for (int q = 0; q < 16; ++q) lds[tid * 17 + q] = a[q];
    __syncthreads();

    // Wave 0 (EXEC all-ones over 32 lanes) reduces 256 partials per quantity
    // with the f32 WMMA accumulator chain: D[q][*] += sum_k A[q][k] * 1.
    if (tid < 32) {
        const int row   = tid & 15;        // quantity index (M dim)
        const int khalf = (tid >> 4) * 2;  // lanes 16-31 hold K=2,3
        v8f d = {};
        v2f ones; ones[0] = 1.0f; ones[1] = 1.0f;
        for (int c = 0; c < kTPB / 4; ++c) {
            v2f av;
            av[0] = lds[(c * 4 + khalf + 0) * 17 + row];   // K = khalf
            av[1] = lds[(c * 4 + khalf + 1) * 17 + row];   // K = khalf+1
            d = __builtin_amdgcn_wmma_f32_16x16x4_f32(
                    false, av, false, ones, (short)0, d, false, false);
        }
        // C/D layout: lane 0 holds D[M=0..7][N=0] in v0..v7, lane 16 -> M=8..15
        if (tid == 0) {
#pragma unroll
            for (int i = 0; i < 8; ++i) part[b * 16 + i] = d[i];
        } else if (tid == 16) {
#pragma unroll
            for (int i = 0; i < 8; ++i) part[b * 16 + 8 + i] = d[i];
        }
    }
}

// ---------------------------------------------------------------------------
// Kernel 2: per-batch 3x3 weighted-Kabsch solve (fp64 internally).
// cov = (M - sm*Q^T - P*cm^T + W*sm*cm^T) / sum_w ; Jacobi on cov^T cov -> V;
// U = cov*V/sigma with cross-product re-orthogonalization; R = V U^T;
// t = cm - R sm.  (3rd-column sign flips are absorbed by the det term.)
// ---------------------------------------------------------------------------
__device__ static inline void cross3(const double* u, const double* v, double* o) {
    o[0] = u[1] * v[2] - u[2] * v[1];
    o[1] = u[2] * v[0] - u[0] * v[2];
    o[2] = u[0] * v[1] - u[1] * v[0];
}

__global__ __launch_bounds__(kTPB) void solve_kernel(
    const float* __restrict__ part, float* __restrict__ out, int B)
{
    const int b = blockIdx.x * blockDim.x + threadIdx.x;
    if (b >= B) return;
    const float* s = part + b * 16;

    const double W = s[0];
    double P[3] = { s[1], s[2], s[3] };
    double Q[3] = { s[4], s[5], s[6] };
    double M[3][3];
#pragma unroll
    for (int i = 0; i < 3; ++i)
#pragma unroll
        for (int j = 0; j < 3; ++j) M[i][j] = s[7 + 3 * i + j];

    const double sum_w = W + (double)kEPS;
    const double denom = W / sum_w + (double)kEPS;
    double sm[3], cm[3];
#pragma unroll
    for (int i = 0; i < 3; ++i) {
        sm[i] = P[i] / (sum_w * denom);
        cm[i] = Q[i] / (sum_w * denom);
    }
    double A[3][3];
#pragma unroll
    for (int i = 0; i < 3; ++i)
#pragma unroll
        for (int j = 0; j < 3; ++j)
            A[i][j] = (M[i][j] - sm[i] * Q[j] - cm[j] * P[i] + sm[i] * cm[j] * W) / sum_w;

    // Bm = A^T A (symmetric)
    double Bm[3][3];
#pragma unroll
    for (int i = 0; i < 3; ++i)
#pragma unroll
        for (int j = 0; j < 3; ++j)
            Bm[i][j] = A[0][i] * A[0][j] + A[1][i] * A[1][j] + A[2][i] * A[2][j];

    double V[3][3] = { {1,0,0},{0,1,0},{0,0,1} };
    const int pq[3][2] = { {0,1},{0,2},{1,2} };
    for (int sweep = 0; sweep < 10; ++sweep) {
        for (int r = 0; r < 3; ++r) {
            const int p = pq[r][0], q = pq[r][1];
            const double apq = Bm[p][q];
            if (fabs(apq) > 1e-300) {
                const double theta = (Bm[q][q] - Bm[p][p]) / (2.0 * apq);
                const double tt = (theta >= 0.0 ? 1.0 : -1.0) /
                                  (fabs(theta) + sqrt(1.0 + theta * theta));
                const double c = 1.0 / sqrt(1.0 + tt * tt);
                const double sn = tt * c;
                for (int k = 0; k < 3; ++k) {
                    const double bkp = Bm[k][p], bkq = Bm[k][q];
                    Bm[k][p] = c * bkp - sn * bkq;
                    Bm[k][q] = sn * bkp + c * bkq;
                }
                for (int k = 0; k < 3; ++k) {
                    const double bpk = Bm[p][k], bqk = Bm[q][k];
                    Bm[p][k] = c * bpk - sn * bqk;
                    Bm[q][k] = sn * bpk + c * bqk;
                }
                for (int k = 0; k < 3; ++k) {
                    const double vkp = V[k][p], vkq = V[k][q];
                    V[k][p] = c * vkp - sn * vkq;
                    V[k][q] = sn * vkp + c * vkq;
                }
            }
        }
    }

    // sort eigenpairs descending
    double ev[3] = { Bm[0][0], Bm[1][1], Bm[2][2] };
    int id[3] = { 0, 1, 2 };
    if (ev[id[0]] < ev[id[1]]) { int t = id[0]; id[0] = id[1]; id[1] = t; }
    if (ev[id[0]] < ev[id[2]]) { int t = id[0]; id[0] = id[2]; id[2] = t; }
    if (ev[id[1]] < ev[id[2]]) { int t = id[1]; id[1] = id[2]; id[2] = t; }

    double Vc[3][3];   // Vc[k] = k-th right singular vector
#pragma unroll
    for (int k = 0; k < 3; ++k)
        for (int i = 0; i < 3; ++i) Vc[k][i] = V[i][id[k]];
    cross3(Vc[0], Vc[1], Vc[2]);   // right-handed V

    double Uc[3][3];
    for (int k = 0; k < 2; ++k) {
        for (int i = 0; i < 3; ++i)
            Uc[k][i] = A[i][0] * Vc[k][0] + A[i][1] * Vc[k][1] + A[i][2] * Vc[k][2];
        if (k == 1) {   // re-orthogonalize against u0
            const double d01 = Uc[1][0] * Uc[0][0] + Uc[1][1] * Uc[0][1] + Uc[1][2] * Uc[0][2];
            for (int i = 0; i < 3; ++i) Uc[1][i] -= d01 * Uc[0][i];
        }
        const double nn = sqrt(Uc[k][0] * Uc[k][0] + Uc[k][1] * Uc[k][1] + Uc[k][2] * Uc[k][2]);
        const double inv = 1.0 / fmax(nn, 1e-300);
        for (int i = 0; i < 3; ++i) Uc[k][i] *= inv;
    }
    cross3(Uc[0], Uc[1], Uc[2]);   // right-handed U

    // R = V U^T  (reflection handling absorbed by forcing det(V)=det(U)=+1)
    double R[3][3];
#pragma unroll
    for (int i = 0; i < 3; ++i)
#pragma unroll
        for (int j = 0; j < 3; ++j)
            R[i][j] = Vc[0][i] * Uc[0][j] + Vc[1][i] * Uc[1][j] + Vc[2][i] * Uc[2][j];

    float* rout = out + (size_t)b * 9;
#pragma unroll
    for (int i = 0; i < 3; ++i)
#pragma unroll
        for (int j = 0; j < 3; ++j)
            rout[i * 3 + j] = (float)R[i][j];

    float* tout = out + (size_t)B * 9 + (size_t)b * 3;
#pragma unroll
    for (int i = 0; i < 3; ++i) {
        const double ti = cm[i] - (R[i][0] * sm[0] + R[i][1] * sm[1] + R[i][2] * sm[2]);
        tout[i] = (float)ti;
    }
}

// ---------------------------------------------------------------------------
extern "C" void kernel_launch(void* const* d_in, const int* in_sizes, int n_in,
                              void* d_out, int out_size, void* d_ws, size_t ws_size,
                              hipStream_t stream)
{
    const float* src = (const float*)d_in[0];
    const float* rcs = (const float*)d_in[1];
    const float* wts = (const float*)d_in[2];
    const int B = in_sizes[2] / kN;          // 1024

    float* part = (float*)d_ws;              // B*16 floats, fully overwritten
    float* outp = (float*)d_out;

    moments_kernel<<<dim3(B), dim3(kTPB), 0, stream>>>(src, rcs, wts, part);
    const int blocks = (B + kTPB - 1) / kTPB;
    solve_kernel<<<dim3(blocks), dim3(kTPB), 0, stream>>>(part, outp, B);
}